// IterNormMod_49177375539586
// MI455X (gfx1250) — compile-verified
//
#include <hip/hip_runtime.h>

typedef __attribute__((ext_vector_type(2))) float v2f;
typedef __attribute__((ext_vector_type(4))) float f4;
typedef __attribute__((ext_vector_type(8))) float v8f;
typedef __attribute__((ext_vector_type(4))) unsigned int u32x4;
typedef __attribute__((ext_vector_type(4))) int i32x4;
typedef __attribute__((ext_vector_type(8))) int i32x8;

#define N_      64
#define C_      256
#define HW_     3136          // 56*56 ; 3136 = 49*64 -> 64-aligned m-slabs never cross n
#define D_      64            // whitening block size
#define G_      4             // groups = 256/64
#define M_      200704        // N_*HW_
#define T_ITERS 10
#define EPS_    1e-5f
#define PAD     65            // LDS row stride (floats), bank-conflict-free

// workspace layout (float offsets)
#define WS_SX   0             // 256   : per-channel sums
#define WS_SXX  256           // 16384 : per-group sum x x^T
#define WS_SIGN 16640         // 16384 : Sigma_N (trace-normalized)
#define WS_WM   33024         // 16384 : whitening matrices

__device__ __forceinline__ v8f wmma4(v2f a, v2f b, v8f c) {
    // D = A(16x4 f32) * B(4x16 f32) + C(16x16 f32)
    return __builtin_amdgcn_wmma_f32_16x16x4_f32(
        false, a, false, b, (short)0, c, false, false);
}

// ---------------------------------------------------------------------------
// TDM: async-load a 64x64 fp32 tile (row stride = row_stride_elems) into LDS.
// D# built per CDNA5 ISA 8.3/8.4.  PAD_EN pads LDS dest by 1 DWORD every
// 64 DWORDs (row stride 65).  Tracked with TENSORcnt.
// ---------------------------------------------------------------------------
template<int CPOL, int PAD_EN>
__device__ __forceinline__ void tdm_load_64x64(const float* gptr,
                                               void* lds_ptr,
                                               int row_stride_elems) {
    unsigned long long ga = (unsigned long long)(size_t)gptr;
    unsigned lds_off = (unsigned)(unsigned long long)(size_t)lds_ptr;

    u32x4 g0;
    g0[0] = 1u;                                        // count=1, user mode
    g0[1] = lds_off;                                   // lds_addr (bytes)
    g0[2] = (unsigned)(ga & 0xFFFFFFFFu);              // global_addr[31:0]
    g0[3] = (unsigned)((ga >> 32) & 0x1FFFFFFu)        // global_addr[56:32]
          | (2u << 30);                                // type=2 ("image")

    unsigned dw0 = (2u << 16);                         // data_size = 4 bytes
    if (PAD_EN) dw0 |= (1u << 20)                      // pad_enable
                    |  (5u << 22);                     // pad_interval: 64 DW
                                                       // pad_amount=0 -> 1 DW
    i32x8 g1;
    g1[0] = (int)dw0;
    g1[1] = (int)(64u << 16);                          // tensor_dim0 = 64 (lo16)
    g1[2] = (int)(64u << 16);                          // tensor_dim1 = 64 (lo16)
    g1[3] = (int)(64u << 16);                          // tile_dim0 = 64
    g1[4] = (int)64;                                   // tile_dim1 = 64, tile_dim2 = 0
    g1[5] = (int)(unsigned)row_stride_elems;           // tensor_dim0_stride lo32
    g1[6] = 0;
    g1[7] = 0;

    i32x4 gz = {0, 0, 0, 0};
#if defined(__clang_major__) && (__clang_major__ >= 23)
    i32x8 gz8 = {0, 0, 0, 0, 0, 0, 0, 0};
    __builtin_amdgcn_tensor_load_to_lds(g0, g1, gz, gz, gz8, CPOL);
#else
    __builtin_amdgcn_tensor_load_to_lds(g0, g1, gz, gz, CPOL);
#endif
}

__global__ void iternorm_zero(float* __restrict__ ws, int n) {
    int i = blockIdx.x * blockDim.x + threadIdx.x;
    if (i < n) ws[i] = 0.0f;
}

// ---------------------------------------------------------------------------
// Pass 1: Sx[c] = sum_m x[c][m],  Sxx[g][d][e] = sum_m x[d][m]*x[e][m]
// grid = (M_/1024, G_), block = 256.  TDM double-buffered slab staging.
// ---------------------------------------------------------------------------
__global__ void __launch_bounds__(256) iternorm_cov(const float* __restrict__ X,
                                                    float* __restrict__ ws) {
    __shared__ float xs[2][D_ * PAD];

    const int gi    = blockIdx.y;
    const int mbase = blockIdx.x * 1024;
    const int tid   = threadIdx.x;
    const int lane  = tid & 31;
    const int wv    = tid >> 5;
    const int lM    = lane & 15;
    const int hs    = lane >> 4;
    const int gbase = gi * D_;

    float* Sx  = ws + WS_SX  + gbase;
    float* Sxx = ws + WS_SXX + gi * D_ * D_;

    const int ti0 = wv * 2, ti1 = wv * 2 + 1;
    const int di0 = ti0 >> 2, ei0 = ti0 & 3;
    const int di1 = ti1 >> 2, ei1 = ti1 & 3;

    v8f acc0 = {}; v8f acc1 = {};
    float csum = 0.0f;

    if (wv == 0) {                                 // prime buffer 0
        const int kb = mbase;
        const int n = kb / HW_, hw = kb - n * HW_;
        tdm_load_64x64<0, 1>(X + (size_t)(n * C_ + gbase) * HW_ + hw,
                             &xs[0][0], HW_);
    }

    for (int s = 0; s < 16; ++s) {                 // 16 slabs of 64 m-values
        if (wv == 0) {
            if (s + 1 < 16) {                      // issue next, then wait cur
                const int kb = mbase + (s + 1) * 64;
                const int n = kb / HW_, hw = kb - n * HW_;
                tdm_load_64x64<0, 1>(X + (size_t)(n * C_ + gbase) * HW_ + hw,
                                     &xs[(s + 1) & 1][0], HW_);
                __builtin_amdgcn_s_wait_tensorcnt(1);
            } else {
                __builtin_amdgcn_s_wait_tensorcnt(0);
            }
        }
        __syncthreads();                           // slab s visible to all
        const float* sb = &xs[s & 1][0];

        if (tid < D_) {                            // channel sums while resident
            float s0 = 0.0f;
            for (int j = 0; j < 64; ++j) s0 += sb[tid * PAD + j];
            csum += s0;
        }

        for (int k0 = 0; k0 < 64; k0 += 4) {
            v2f a, b;
            a.x = sb[(di0 * 16 + lM) * PAD + k0 + 2 * hs];
            a.y = sb[(di0 * 16 + lM) * PAD + k0 + 2 * hs + 1];
            b.x = sb[(ei0 * 16 + lM) * PAD + k0 + 2 * hs];
            b.y = sb[(ei0 * 16 + lM) * PAD + k0 + 2 * hs + 1];
            acc0 = wmma4(a, b, acc0);
            a.x = sb[(di1 * 16 + lM) * PAD + k0 + 2 * hs];
            a.y = sb[(di1 * 16 + lM) * PAD + k0 + 2 * hs + 1];
            b.x = sb[(ei1 * 16 + lM) * PAD + k0 + 2 * hs];
            b.y = sb[(ei1 * 16 + lM) * PAD + k0 + 2 * hs + 1];
            acc1 = wmma4(a, b, acc1);
        }
        __syncthreads();                           // done with buf before DMA reuse
    }

    if (tid < D_) atomicAdd(&Sx[tid], csum);
    for (int r = 0; r < 8; ++r) {
        atomicAdd(&Sxx[(di0 * 16 + r + 8 * hs) * D_ + ei0 * 16 + lM], acc0[r]);
        atomicAdd(&Sxx[(di1 * 16 + r + 8 * hs) * D_ + ei1 * 16 + lM], acc1[r]);
    }
}

// ---------------------------------------------------------------------------
// 64x64x64 matmul tile helpers (Dst/A/B in LDS stride PAD; Bg global)
// ---------------------------------------------------------------------------
__device__ __forceinline__ void mm_tile_lds(float* Dst, const float* A,
                                            const float* B, int di, int ei,
                                            int lM, int hs) {
    v8f acc = {};
    for (int k0 = 0; k0 < 64; k0 += 4) {
        v2f a, b;
        a.x = A[(di * 16 + lM) * PAD + k0 + 2 * hs];
        a.y = A[(di * 16 + lM) * PAD + k0 + 2 * hs + 1];
        b.x = B[(k0 + 2 * hs)     * PAD + ei * 16 + lM];
        b.y = B[(k0 + 2 * hs + 1) * PAD + ei * 16 + lM];
        acc = wmma4(a, b, acc);
    }
    for (int r = 0; r < 8; ++r)
        Dst[(di * 16 + r + 8 * hs) * PAD + ei * 16 + lM] = acc[r];
}

__device__ __forceinline__ void mm_tile_glb(float* Dst, const float* A,
                                            const float* __restrict__ Bg,
                                            int di, int ei, int lM, int hs) {
    v8f acc = {};
    for (int k0 = 0; k0 < 64; k0 += 4) {
        v2f a, b;
        a.x = A[(di * 16 + lM) * PAD + k0 + 2 * hs];
        a.y = A[(di * 16 + lM) * PAD + k0 + 2 * hs + 1];
        b.x = Bg[(k0 + 2 * hs)     * D_ + ei * 16 + lM];
        b.y = Bg[(k0 + 2 * hs + 1) * D_ + ei * 16 + lM];
        acc = wmma4(a, b, acc);
    }
    for (int r = 0; r < 8; ++r)
        Dst[(di * 16 + r + 8 * hs) * PAD + ei * 16 + lM] = acc[r];
}

// ---------------------------------------------------------------------------
// Pass 2: Sigma_N, Newton-Schulz x10, wm = P*sqrt(rTr).  grid = G_, block 256
// ---------------------------------------------------------------------------
__global__ void __launch_bounds__(256) iternorm_newton(float* __restrict__ ws) {
    __shared__ float P [D_ * PAD];
    __shared__ float T1[D_ * PAD];
    __shared__ float T2[D_ * PAD];

    const int gi   = blockIdx.x;
    const int tid  = threadIdx.x;
    const int lane = tid & 31;
    const int wv   = tid >> 5;
    const int lM   = lane & 15;
    const int hs   = lane >> 4;
    const float inv_m = 1.0f / (float)M_;

    const float* Sx  = ws + WS_SX  + gi * D_;
    const float* Sxx = ws + WS_SXX + gi * D_ * D_;
    float*       Sg  = ws + WS_SIGN + gi * D_ * D_;
    float*       WM  = ws + WS_WM  + gi * D_ * D_;

    float tr = 0.0f;                               // trace(Sigma), all lanes
    for (int dd = 0; dd < D_; ++dd) {
        float mu = Sx[dd] * inv_m;
        tr += EPS_ + Sxx[dd * D_ + dd] * inv_m - mu * mu;
    }
    const float rTr = 1.0f / tr;

    for (int i = tid; i < D_ * D_; i += 256) {
        int r = i >> 6, c = i & 63;
        float mur = Sx[r] * inv_m, muc = Sx[c] * inv_m;
        float sig = Sxx[i] * inv_m - mur * muc + ((r == c) ? EPS_ : 0.0f);
        Sg[i] = sig * rTr;                          // Sigma_N (global, L2-hot)
        P[r * PAD + c] = (r == c) ? 1.0f : 0.0f;    // P0 = I
    }
    __threadfence();
    __syncthreads();

    const int ti0 = wv * 2, ti1 = ti0 + 1;
    const int di0 = ti0 >> 2, ei0 = ti0 & 3;
    const int di1 = ti1 >> 2, ei1 = ti1 & 3;

    for (int it = 0; it < T_ITERS; ++it) {
        mm_tile_lds(T1, P, P, di0, ei0, lM, hs);    // T1 = P*P
        mm_tile_lds(T1, P, P, di1, ei1, lM, hs);
        __syncthreads();
        mm_tile_lds(T2, T1, P, di0, ei0, lM, hs);   // T2 = P^3
        mm_tile_lds(T2, T1, P, di1, ei1, lM, hs);
        __syncthreads();
        mm_tile_glb(T1, T2, Sg, di0, ei0, lM, hs);  // T1 = P^3 * Sigma_N
        mm_tile_glb(T1, T2, Sg, di1, ei1, lM, hs);
        __syncthreads();
        for (int i = tid; i < D_ * D_; i += 256) {
            int r = i >> 6, c = i & 63;
            P[r * PAD + c] = 1.5f * P[r * PAD + c] - 0.5f * T1[r * PAD + c];
        }
        __syncthreads();
    }

    const float srt = sqrtf(rTr);
    for (int i = tid; i < D_ * D_; i += 256) {
        int r = i >> 6, c = i & 63;
        WM[i] = P[r * PAD + c] * srt;
    }
}

// ---------------------------------------------------------------------------
// Pass 3: out = (wm @ x - wm @ mu) * weight + bias
// grid = (M_/64, G_), block = 256.  TDM x-slab, LDS-staged NT float4 stores.
// ---------------------------------------------------------------------------
__global__ void __launch_bounds__(256) iternorm_apply(
        const float* __restrict__ X, const float* __restrict__ weight,
        const float* __restrict__ bias, const float* __restrict__ ws,
        float* __restrict__ out) {
    __shared__ float wmS [D_ * PAD];
    __shared__ float xsS [D_ * 64];                 // stride 64: col-reads only
    __shared__ float outS[D_ * 64];
    __shared__ float off [D_];

    const int gi    = blockIdx.y;
    const int mbase = blockIdx.x * 64;
    const int tid   = threadIdx.x;
    const int lane  = tid & 31;
    const int wv    = tid >> 5;
    const int lM    = lane & 15;
    const int hs    = lane >> 4;
    const int gbase = gi * D_;
    const float inv_m = 1.0f / (float)M_;

    const int n   = mbase / HW_;                    // slab never crosses n
    const int hw0 = mbase - n * HW_;

    const float* Sx = ws + WS_SX + gbase;
    const float* WM = ws + WS_WM + gi * D_ * D_;

    if (wv == 0)                                    // NT: last reader of X
        tdm_load_64x64<1, 0>(X + (size_t)(n * C_ + gbase) * HW_ + hw0,
                             &xsS[0], HW_);

    for (int i = tid; i < D_ * D_; i += 256)
        wmS[(i >> 6) * PAD + (i & 63)] = WM[i];
    __syncthreads();                                // wmS visible
    if (tid < D_) {                                 // off = wm @ mean
        float o = 0.0f;
        for (int e = 0; e < D_; ++e) o += wmS[tid * PAD + e] * (Sx[e] * inv_m);
        off[tid] = o;
    }
    if (wv == 0) __builtin_amdgcn_s_wait_tensorcnt(0);
    __syncthreads();                                // xsS + off visible

    for (int t = 0; t < 2; ++t) {
        const int ti = wv * 2 + t;
        const int di = ti >> 2, ei = ti & 3;        // di: row block, ei: m block
        v8f acc = {};
        for (int k0 = 0; k0 < 64; k0 += 4) {
            v2f a, b;
            a.x = wmS[(di * 16 + lM) * PAD + k0 + 2 * hs];
            a.y = wmS[(di * 16 + lM) * PAD + k0 + 2 * hs + 1];
            b.x = xsS[(k0 + 2 * hs)     * 64 + ei * 16 + lM];
            b.y = xsS[(k0 + 2 * hs + 1) * 64 + ei * 16 + lM];
            acc = wmma4(a, b, acc);
        }
        for (int r = 0; r < 8; ++r) {               // stage with fused affine
            int row = di * 16 + r + 8 * hs;
            int c   = gbase + row;
            outS[row * 64 + ei * 16 + lM] =
                (acc[r] - off[row]) * weight[c] + bias[c];
        }
    }
    __syncthreads();

    for (int i = tid; i < 1024; i += 256) {         // 256B/row NT streams
        int row = i >> 4;
        int j   = (i & 15) * 4;
        f4 v = *(const f4*)&outS[row * 64 + j];
        __builtin_nontemporal_store(
            v, (f4*)(out + (size_t)(n * C_ + gbase + row) * HW_ + hw0 + j));
    }
}

extern "C" void kernel_launch(void* const* d_in, const int* in_sizes, int n_in,
                              void* d_out, int out_size, void* d_ws, size_t ws_size,
                              hipStream_t stream) {
    const float* X      = (const float*)d_in[0];
    const float* weight = (const float*)d_in[1];
    const float* bias   = (const float*)d_in[2];
    float* out = (float*)d_out;
    float* ws  = (float*)d_ws;

    iternorm_zero<<<(16640 + 255) / 256, 256, 0, stream>>>(ws, 16640);

    dim3 gcov(M_ / 1024, G_);
    iternorm_cov<<<gcov, 256, 0, stream>>>(X, ws);

    iternorm_newton<<<G_, 256, 0, stream>>>(ws);

    dim3 gapp(M_ / 64, G_);
    iternorm_apply<<<gapp, 256, 0, stream>>>(X, weight, bias, ws, out);
}